// RQSLayer_79697413144883
// MI455X (gfx1250) — compile-verified
//
#include <hip/hip_runtime.h>
#include <hip/hip_bf16.h>
#include <stdint.h>

// RQS flow layer for MI455X (gfx1250).
// Two-pass: (1) deterministic reduction B = mean(|theta[:,25]|),
// (2) streaming per-row spline with async global->LDS staging (ASYNCcnt path).

#define TPB  256
#define COLS 26   // 3*BINS + 2
#define KB   8    // BINS

// ---------------- Pass 1: per-block partial sums of |theta[:,25]| ------------
__global__ void __launch_bounds__(TPB)
rqs_reduce1(const float* __restrict__ theta, float* __restrict__ partial, int n)
{
    __shared__ float red[TPB];
    float acc = 0.0f;
    for (int r = blockIdx.x * TPB + threadIdx.x; r < n; r += gridDim.x * TPB)
        acc += fabsf(theta[(size_t)r * COLS + (COLS - 1)]);
    red[threadIdx.x] = acc;
    __syncthreads();
    #pragma unroll
    for (int s = TPB / 2; s > 0; s >>= 1) {
        if (threadIdx.x < s) red[threadIdx.x] += red[threadIdx.x + s];
        __syncthreads();
    }
    if (threadIdx.x == 0) partial[blockIdx.x] = red[0];
}

// ---------------- Pass 1b: finalize B -> ws[0] -------------------------------
__global__ void __launch_bounds__(TPB)
rqs_reduce2(const float* __restrict__ partial, float* __restrict__ Bslot, int n)
{
    __shared__ float red[TPB];
    float acc = partial[threadIdx.x] + partial[threadIdx.x + 256]
              + partial[threadIdx.x + 512] + partial[threadIdx.x + 768];
    red[threadIdx.x] = acc;
    __syncthreads();
    #pragma unroll
    for (int s = TPB / 2; s > 0; s >>= 1) {
        if (threadIdx.x < s) red[threadIdx.x] += red[threadIdx.x + s];
        __syncthreads();
    }
    if (threadIdx.x == 0) Bslot[0] = red[0] / (float)n;
}

// ---------------- Per-row spline math (all registers, fully unrolled) --------
__device__ __forceinline__ void rqs_row(const float th[COLS], float x, float B,
                                        float& out, float& ld)
{
    const float MIN_W = 0.001f, MIN_H = 0.001f, MIN_D = 0.001f;

    // softmax over widths logits
    float mw = th[0];
    #pragma unroll
    for (int i = 1; i < KB; ++i) mw = fmaxf(mw, th[i]);
    float ew[KB]; float sw = 0.0f;
    #pragma unroll
    for (int i = 0; i < KB; ++i) { ew[i] = expf(th[i] - mw); sw += ew[i]; }
    float fw = (1.0f - (float)KB * MIN_W) / sw;

    // softmax over heights logits
    float mh = th[KB];
    #pragma unroll
    for (int i = 1; i < KB; ++i) mh = fmaxf(mh, th[KB + i]);
    float eh[KB]; float sh = 0.0f;
    #pragma unroll
    for (int i = 0; i < KB; ++i) { eh[i] = expf(th[KB + i] - mh); sh += eh[i]; }
    float fh = (1.0f - (float)KB * MIN_H) / sh;

    // knot positions (affine-mapped cumsums, endpoints pinned to +/-B)
    float cw[KB + 1], ch[KB + 1];
    cw[0] = -B; ch[0] = -B;
    float rw = 0.0f, rh = 0.0f;
    #pragma unroll
    for (int i = 0; i < KB; ++i) {
        rw += MIN_W + fw * ew[i];
        rh += MIN_H + fh * eh[i];
        cw[i + 1] = 2.0f * B * rw - B;
        ch[i + 1] = 2.0f * B * rh - B;
    }
    cw[KB] = B; ch[KB] = B;

    // derivatives: MIN_D + softplus(ud), stable form
    float dv[KB + 1];
    #pragma unroll
    for (int i = 0; i <= KB; ++i) {
        float u = th[2 * KB + i];
        dv[i] = MIN_D + fmaxf(u, 0.0f) + log1pf(expf(-fabsf(u)));
    }

    bool  inside = (x >= -B) && (x <= B);
    float xs = fminf(fmaxf(x, -B), B);

    // bin search + gather as an unrolled select chain (stays in VGPRs)
    float cwlo = cw[0], cwhi = cw[1];
    float chlo = ch[0], chhi = ch[1];
    float d0 = dv[0], d1 = dv[1];
    #pragma unroll
    for (int j = 1; j < KB; ++j) {
        bool s = (xs >= cw[j]);
        cwlo = s ? cw[j]     : cwlo;
        cwhi = s ? cw[j + 1] : cwhi;
        chlo = s ? ch[j]     : chlo;
        chhi = s ? ch[j + 1] : chhi;
        d0   = s ? dv[j]     : d0;
        d1   = s ? dv[j + 1] : d1;
    }

    float in_w  = cwhi - cwlo;
    float in_h  = chhi - chlo;
    float delta = in_h / in_w;
    float t     = (xs - cwlo) / in_w;
    float omt   = 1.0f - t;
    float t1mt  = t * omt;
    float numer = in_h * (delta * t * t + d0 * t1mt);
    float denom = delta + (d0 + d1 - 2.0f * delta) * t1mt;
    float spl   = chlo + numer / denom;
    float dnum  = delta * delta * (d1 * t * t + 2.0f * delta * t1mt + d0 * omt * omt);
    float lad   = logf(dnum) - 2.0f * logf(denom);

    out = inside ? spl : x;
    ld  = inside ? lad : 0.0f;
}

// ---------------- Pass 2: main streaming kernel ------------------------------
__global__ void __launch_bounds__(TPB)
rqs_main(const float* __restrict__ theta, const float* __restrict__ y,
         const float* __restrict__ Bslot, float* __restrict__ out, int n)
{
    __shared__ float smem[TPB * COLS];          // 26624 B per block
    const int t    = threadIdx.x;
    const int base = blockIdx.x * TPB;
    const int r    = base + t;
    const float B  = Bslot[0];

    float th[COLS];

    if (base + TPB <= n) {
        // Full block: stage 256 rows (26624 B) of theta into LDS with the
        // gfx1250 async copy engine. 6x b128 + 1x b64 per lane, coalesced.
        const float* gsrc = theta + (size_t)base * COLS;
        unsigned smemBase = (unsigned)(uintptr_t)(&smem[0]); // low 32b = LDS offset
        #pragma unroll
        for (int i = 0; i < 6; ++i) {
            unsigned e   = (unsigned)((i * TPB + t) * 16);   // byte offset
            unsigned lds = smemBase + e;
            asm volatile("global_load_async_to_lds_b128 %0, %1, %2"
                         :: "v"(lds), "v"(e), "s"(gsrc) : "memory");
        }
        {
            unsigned e   = 24576u + (unsigned)t * 8u;        // remaining 2048 B
            unsigned lds = smemBase + e;
            asm volatile("global_load_async_to_lds_b64 %0, %1, %2"
                         :: "v"(lds), "v"(e), "s"(gsrc) : "memory");
        }
        asm volatile("s_wait_asynccnt 0" ::: "memory");
        __syncthreads();
        // stride-104B row reads: conflict-free across 64 LDS banks
        #pragma unroll
        for (int i = 0; i < COLS; ++i) th[i] = smem[t * COLS + i];
    } else {
        // Tail block: guarded direct loads
        if (r < n) {
            const float* g = theta + (size_t)r * COLS;
            #pragma unroll
            for (int i = 0; i < COLS; ++i) th[i] = g[i];
        } else {
            #pragma unroll
            for (int i = 0; i < COLS; ++i) th[i] = 0.0f;
        }
    }

    if (r >= n) return;

    float x = y[r];
    float Tu, ld;
    rqs_row(th, x, B, Tu, ld);

    out[r]     = Tu;   // Tu (N,1)
    out[n + r] = ld;   // logabsdet (N,)
}

// ---------------- Launch -----------------------------------------------------
extern "C" void kernel_launch(void* const* d_in, const int* in_sizes, int n_in,
                              void* d_out, int out_size, void* d_ws, size_t ws_size,
                              hipStream_t stream)
{
    const float* theta = (const float*)d_in[0];   // (N, 26) f32
    const float* y     = (const float*)d_in[1];   // (N, 1)  f32
    const int n = in_sizes[1];                    // N = 2,000,000

    float* ws      = (float*)d_ws;
    float* Bslot   = ws;        // ws[0]        : B
    float* partial = ws + 1;    // ws[1..1024]  : block partials (4100 B total)

    rqs_reduce1<<<1024, TPB, 0, stream>>>(theta, partial, n);
    rqs_reduce2<<<1,    TPB, 0, stream>>>(partial, Bslot, n);

    const int grid = (n + TPB - 1) / TPB;
    rqs_main<<<grid, TPB, 0, stream>>>(theta, y, Bslot, (float*)d_out, n);
}